// DKVMN_82970178224636
// MI455X (gfx1250) — compile-verified
//
#include <hip/hip_runtime.h>
#include <hip/hip_bf16.h>

// ---------------- problem constants ----------------
#define NQ   10000      // num questions
#define DD   64         // dim
#define MM   50         // memory slots
#define BB   64         // batch
#define TT   512        // seq len
#define T1   511        // T-1
#define NROW (BB*TT)    // 32768 rows for front GEMMs
#define N2   (BB*T1)    // 32704 rows for output GEMM
#define LDSW 68         // padded LDS row stride (floats) -> conflict-free WMMA frags
#define GST  192        // per-step interleaved operand stride (floats): w@0,e@64,a@128

typedef __attribute__((ext_vector_type(2))) float v2f;
typedef __attribute__((ext_vector_type(8))) float v8f;

__device__ __forceinline__ v8f v8f_zero() {
    v8f z;
#pragma unroll
    for (int i = 0; i < 8; ++i) z[i] = 0.0f;
    return z;
}

// 16x16 output tile, K-dim loop in steps of 4, full-f32 WMMA.
// A[r][k] at As[r*lda + k]   (row tile base already applied)
// B[k][n] at Bs[k*ldb + n]   (col tile base already applied)
__device__ __forceinline__ v8f wmma16x16_f32(const float* As, int lda,
                                             const float* Bs, int ldb,
                                             int K, v8f c) {
    const int lane = threadIdx.x & 31;
    const int half = lane >> 4;   // 0: K pair {k0,k0+1}, 1: {k0+2,k0+3}
    const int l16  = lane & 15;   // M index for A, N index for B
#pragma unroll
    for (int k0 = 0; k0 < K; k0 += 4) {
        v2f a, b;
        const float* ap = As + l16 * lda + k0 + 2 * half;
        a.x = ap[0];
        a.y = ap[1];
        const float* bp = Bs + (k0 + 2 * half) * ldb + l16;
        b.x = bp[0];
        b.y = bp[ldb];
        c = __builtin_amdgcn_wmma_f32_16x16x4_f32(false, a, false, b,
                                                  (short)0, c, false, false);
    }
    return c;
}

// ============================================================
// Phase A: gather + w-logits GEMM + softmax, e GEMM + sigmoid,
//          a GEMM + tanh.  One block = 64 rows of [B*T].
// Results stored interleaved: gBuf[row*192 + {0..49:w, 64..127:e, 128..191:a}]
// ============================================================
__global__ __launch_bounds__(256)
void dkvmn_front(const int* __restrict__ question, const int* __restrict__ response,
                 const float* __restrict__ k_emb, const float* __restrict__ v_emb,
                 const float* __restrict__ Mk,
                 const float* __restrict__ e_W, const float* __restrict__ e_b,
                 const float* __restrict__ a_W, const float* __restrict__ a_b,
                 float* __restrict__ gBuf) {
    __shared__ float bufA[64 * LDSW];   // activations (k rows, then v rows)
    __shared__ float bufB[64 * LDSW];   // weights (MkT, then e_W, then a_W)
    __shared__ float bufO[64 * LDSW];   // w logits

    const int tid  = threadIdx.x;
    const int wave = tid >> 5;
    const int lane = tid & 31;
    const int half = lane >> 4;
    const int l16  = lane & 15;
    const int R0   = blockIdx.x * 64;

    // ---- stage gathered k rows ----
    for (int i = tid; i < 64 * 16; i += 256) {
        int row = i >> 4, seg = i & 15;
        int q = question[R0 + row];
        float4 v = reinterpret_cast<const float4*>(k_emb + q * DD)[seg];
        *reinterpret_cast<float4*>(&bufA[row * LDSW + seg * 4]) = v;
    }
    // ---- stage Mk^T (padded: cols >= 50 are zero) ----
    for (int i = tid; i < 64 * 64; i += 256) {
        int d = i >> 6, m = i & 63;
        bufB[d * LDSW + m] = (m < MM) ? Mk[m * DD + d] : 0.0f;
    }
    __syncthreads();

    // ---- GEMM1: logits = K @ Mk^T ----
    for (int tile = wave; tile < 16; tile += 8) {
        int rt = tile >> 2, ct = tile & 3;
        v8f c = wmma16x16_f32(&bufA[rt * 16 * LDSW], LDSW,
                              &bufB[ct * 16], LDSW, DD, v8f_zero());
#pragma unroll
        for (int i = 0; i < 8; ++i)
            bufO[(rt * 16 + i + 8 * half) * LDSW + ct * 16 + l16] = c[i];
    }
    __syncthreads();

    // ---- row softmax over first 50 cols -> gBuf ----
    if (tid < 64) {
        float mx = -3.0e38f;
        for (int m = 0; m < MM; ++m) mx = fmaxf(mx, bufO[tid * LDSW + m]);
        float s = 0.0f;
        for (int m = 0; m < MM; ++m) {
            float ev = expf(bufO[tid * LDSW + m] - mx);
            s += ev;
            bufO[tid * LDSW + m] = ev;
        }
        float inv = 1.0f / s;
        for (int m = 0; m < MM; ++m)
            gBuf[(size_t)(R0 + tid) * GST + m] = bufO[tid * LDSW + m] * inv;
    }

    // ---- restage: gathered v rows and e_W (safe: GEMM1 reads done) ----
    for (int i = tid; i < 64 * 16; i += 256) {
        int row = i >> 4, seg = i & 15;
        int x = question[R0 + row] + NQ * response[R0 + row];
        float4 v = reinterpret_cast<const float4*>(v_emb + (size_t)x * DD)[seg];
        *reinterpret_cast<float4*>(&bufA[row * LDSW + seg * 4]) = v;
    }
    for (int i = tid; i < 64 * 16; i += 256) {
        int row = i >> 4, seg = i & 15;
        float4 v = reinterpret_cast<const float4*>(e_W + row * DD)[seg];
        *reinterpret_cast<float4*>(&bufB[row * LDSW + seg * 4]) = v;
    }
    __syncthreads();

    // ---- GEMM2: e = sigmoid(V @ e_W + e_b) ----
    for (int tile = wave; tile < 16; tile += 8) {
        int rt = tile >> 2, ct = tile & 3;
        v8f c = wmma16x16_f32(&bufA[rt * 16 * LDSW], LDSW,
                              &bufB[ct * 16], LDSW, DD, v8f_zero());
        int col = ct * 16 + l16;
        float bias = e_b[col];
#pragma unroll
        for (int i = 0; i < 8; ++i) {
            int row = R0 + rt * 16 + i + 8 * half;
            gBuf[(size_t)row * GST + 64 + col] = 1.0f / (1.0f + expf(-(c[i] + bias)));
        }
    }
    __syncthreads();

    // ---- restage a_W over bufB ----
    for (int i = tid; i < 64 * 16; i += 256) {
        int row = i >> 4, seg = i & 15;
        float4 v = reinterpret_cast<const float4*>(a_W + row * DD)[seg];
        *reinterpret_cast<float4*>(&bufB[row * LDSW + seg * 4]) = v;
    }
    __syncthreads();

    // ---- GEMM3: a = tanh(V @ a_W + a_b) ----
    for (int tile = wave; tile < 16; tile += 8) {
        int rt = tile >> 2, ct = tile & 3;
        v8f c = wmma16x16_f32(&bufA[rt * 16 * LDSW], LDSW,
                              &bufB[ct * 16], LDSW, DD, v8f_zero());
        int col = ct * 16 + l16;
        float bias = a_b[col];
#pragma unroll
        for (int i = 0; i < 8; ++i) {
            int row = R0 + rt * 16 + i + 8 * half;
            gBuf[(size_t)row * GST + 128 + col] = tanhf(c[i] + bias);
        }
    }
}

// ============================================================
// Phase B: sequential memory scan, one block per batch element.
// Per-step operands double-buffered in LDS via gfx1250 async
// global->LDS copies (ASYNCcnt): issue t+1 before waiting for t,
// so global latency hides behind the previous step's compute.
// ============================================================
__global__ __launch_bounds__(256)
void dkvmn_scan(const float* __restrict__ mask, const float* __restrict__ Mv0,
                const float* __restrict__ gBuf, float* __restrict__ readBuf) {
    __shared__ float Mv[MM * DD];                         // 3200 floats
    __shared__ __align__(16) float stage[2][GST];         // w|e|a double buffer
    __shared__ float red[4 * DD];
    __shared__ float maskS[TT];

    const int b = blockIdx.x;
    const int tid = threadIdx.x;

    const float* gBase = gBuf + (size_t)b * TT * GST;     // uniform per block
    const unsigned ldsStage0 = (unsigned)(uintptr_t)&stage[0][0];
    const unsigned ldsStage1 = (unsigned)(uintptr_t)&stage[1][0];

    for (int i = tid; i < MM * DD; i += 256) Mv[i] = Mv0[i];
    for (int i = tid; i < TT; i += 256) maskS[i] = mask[b * TT + i];

    // prologue: async-stage t=0 into stage[0] (48 lanes x b128 = 192 floats)
    if (tid < 48) {
        unsigned dst = ldsStage0 + tid * 16;
        unsigned off = tid * 16;
        asm volatile("global_load_async_to_lds_b128 %0, %1, %2"
                     :: "v"(dst), "v"(off), "s"(gBase) : "memory");
    }
    __syncthreads();

    for (int t = 0; t < TT; ++t) {
        const int cur = t & 1;
        if (t + 1 < TT) {
            // issue next step into the other buffer, THEN wait for current:
            // ASYNCcnt completes in order, so waiting to <=1 drains step t.
            if (tid < 48) {
                unsigned dst = (cur ? ldsStage0 : ldsStage1) + tid * 16;
                unsigned off = (unsigned)(t + 1) * (GST * 4u) + tid * 16;
                asm volatile("global_load_async_to_lds_b128 %0, %1, %2"
                             :: "v"(dst), "v"(off), "s"(gBase) : "memory");
            }
            asm volatile("s_wait_asynccnt 0x1" ::: "memory");
        } else {
            asm volatile("s_wait_asynccnt 0x0" ::: "memory");
        }
        __syncthreads();   // stage[cur] visible to all waves

        const float* wT = stage[cur];
        const float* eT = stage[cur] + 64;
        const float* aT = stage[cur] + 128;

        if (t > 0) {   // read with state after step t-1 (before update)
            const int d = tid & 63, g = tid >> 6;
            float s = 0.0f;
            for (int m = g; m < MM; m += 4) s += wT[m] * Mv[m * DD + d];
            red[g * DD + d] = s;
            __syncthreads();
            if (g == 0) {
                float r = red[d] + red[DD + d] + red[2 * DD + d] + red[3 * DD + d];
                readBuf[(size_t)(b * T1 + (t - 1)) * DD + d] = r;
            }
            __syncthreads();   // Mv reads complete before update
        }

        if (maskS[t] == 1.0f) {
            for (int i = tid; i < MM * DD; i += 256) {
                int d = i & 63;
                float wv = wT[i >> 6];
                Mv[i] = Mv[i] * (1.0f - wv * eT[d]) + wv * aT[d];
            }
        }
        __syncthreads();   // all reads of stage[cur] done before next issue reuses it
    }
}

// ============================================================
// Phase C: f = tanh([read | k1] @ f_W + f_b); p = f @ p_W + p_b.
// K=128 done as two K=64 WMMA passes reusing LDS buffers.
// ============================================================
__global__ __launch_bounds__(256)
void dkvmn_out(const int* __restrict__ question, const float* __restrict__ k_emb,
               const float* __restrict__ f_W, const float* __restrict__ f_b,
               const float* __restrict__ p_W, const float* __restrict__ p_b,
               const float* __restrict__ readBuf, float* __restrict__ out) {
    __shared__ float bufA[64 * LDSW];
    __shared__ float bufB[64 * LDSW];

    const int tid  = threadIdx.x;
    const int wave = tid >> 5;
    const int lane = tid & 31;
    const int half = lane >> 4;
    const int l16  = lane & 15;
    const int G0   = blockIdx.x * 64;

    // ---- pass 1: read part, f_W rows 0..63 ----
    for (int i = tid; i < 64 * 16; i += 256) {
        int row = i >> 4, seg = i & 15;
        float4 v = reinterpret_cast<const float4*>(readBuf + (size_t)(G0 + row) * DD)[seg];
        *reinterpret_cast<float4*>(&bufA[row * LDSW + seg * 4]) = v;
    }
    for (int i = tid; i < 64 * 16; i += 256) {
        int row = i >> 4, seg = i & 15;
        float4 v = reinterpret_cast<const float4*>(f_W + row * DD)[seg];
        *reinterpret_cast<float4*>(&bufB[row * LDSW + seg * 4]) = v;
    }
    __syncthreads();

    v8f acc[2];
    acc[0] = v8f_zero();
    acc[1] = v8f_zero();
    int tc = 0;
    for (int tile = wave; tile < 16; tile += 8, ++tc) {
        int rt = tile >> 2, ct = tile & 3;
        acc[tc] = wmma16x16_f32(&bufA[rt * 16 * LDSW], LDSW,
                                &bufB[ct * 16], LDSW, DD, acc[tc]);
    }
    __syncthreads();

    // ---- pass 2: k[:,1:] part (re-gather), f_W rows 64..127 ----
    for (int i = tid; i < 64 * 16; i += 256) {
        int row = i >> 4, seg = i & 15;
        int g = G0 + row;
        int b = g / T1;
        int tt = g - b * T1;         // t = tt + 1
        int q = question[b * TT + tt + 1];
        float4 v = reinterpret_cast<const float4*>(k_emb + q * DD)[seg];
        *reinterpret_cast<float4*>(&bufA[row * LDSW + seg * 4]) = v;
    }
    for (int i = tid; i < 64 * 16; i += 256) {
        int row = i >> 4, seg = i & 15;
        float4 v = reinterpret_cast<const float4*>(f_W + (64 + row) * DD)[seg];
        *reinterpret_cast<float4*>(&bufB[row * LDSW + seg * 4]) = v;
    }
    __syncthreads();

    tc = 0;
    for (int tile = wave; tile < 16; tile += 8, ++tc) {
        int rt = tile >> 2, ct = tile & 3;
        acc[tc] = wmma16x16_f32(&bufA[rt * 16 * LDSW], LDSW,
                                &bufB[ct * 16], LDSW, DD, acc[tc]);
    }
    __syncthreads();   // all WMMA LDS reads done; reuse bufA for f

    tc = 0;
    for (int tile = wave; tile < 16; tile += 8, ++tc) {
        int rt = tile >> 2, ct = tile & 3;
        int col = ct * 16 + l16;
        float bias = f_b[col];
#pragma unroll
        for (int i = 0; i < 8; ++i) {
            int row = rt * 16 + i + 8 * half;
            bufA[row * LDSW + col] = tanhf(acc[tc][i] + bias);
        }
    }
    __syncthreads();

    if (tid < 64) {
        float s = p_b[0];
        for (int j = 0; j < DD; ++j) s += bufA[tid * LDSW + j] * p_W[j];
        out[G0 + tid] = s;
    }
}

// ============================================================
extern "C" void kernel_launch(void* const* d_in, const int* in_sizes, int n_in,
                              void* d_out, int out_size, void* d_ws, size_t ws_size,
                              hipStream_t stream) {
    const int*   question = (const int*)d_in[0];
    const int*   response = (const int*)d_in[1];
    const float* mask     = (const float*)d_in[2];
    const float* k_emb    = (const float*)d_in[3];
    const float* v_emb    = (const float*)d_in[4];
    const float* Mk       = (const float*)d_in[5];
    const float* Mv0      = (const float*)d_in[6];
    const float* e_W      = (const float*)d_in[7];
    const float* e_b      = (const float*)d_in[8];
    const float* a_W      = (const float*)d_in[9];
    const float* a_b      = (const float*)d_in[10];
    const float* f_W      = (const float*)d_in[11];
    const float* f_b      = (const float*)d_in[12];
    const float* p_W      = (const float*)d_in[13];
    const float* p_b      = (const float*)d_in[14];
    (void)in_sizes; (void)n_in; (void)out_size; (void)ws_size;

    float* ws      = (float*)d_ws;
    float* gBuf    = ws;                                   // [32768, 192]: w|e|a interleaved
    float* readBuf = gBuf + (size_t)NROW * GST;            // [32704, 64]

    dkvmn_front<<<NROW / 64, 256, 0, stream>>>(question, response, k_emb, v_emb, Mk,
                                               e_W, e_b, a_W, a_b, gBuf);
    dkvmn_scan<<<BB, 256, 0, stream>>>(mask, Mv0, gBuf, readBuf);
    dkvmn_out<<<N2 / 64, 256, 0, stream>>>(question, k_emb, f_W, f_b, p_W, p_b,
                                           readBuf, (float*)d_out);
}